// EmbeddingModel_41961830481956
// MI455X (gfx1250) — compile-verified
//
#include <hip/hip_runtime.h>

// Ragged segment-mean (sorted word_ids) as WMMA ownership-matmul on gfx1250.
// D_tile[16w x 16d] += O[16w x 4k] * E[4k x 16d], with 1/max(count,1) folded
// into the ownership matrix A so the f32 accumulator holds means directly.
//
// Memory policy (MI455X, 23.3 TB/s HBM, 192 MB L2): emb (201 MB, one-touch)
// and out (201 MB, write-once) use non-temporal hints so the streaming data
// does not thrash L2; word_ids (256 KB total, read by 256 blocks/sentence)
// stays on the default temporal path and lives in L2.

typedef __attribute__((ext_vector_type(2))) float v2f;
typedef __attribute__((ext_vector_type(8))) float v8f;

static constexpr int Bn = 16;
static constexpr int Tn = 4096;
static constexpr int Dn = 768;
static constexpr int Wn = 4096;         // word slots per sentence
static constexpr int WPB = 16;          // words per block (one WMMA M-tile)
static constexpr int DPW = 96;          // dims per wave (6 N-tiles of 16)

__device__ __forceinline__ int lbound4096(const int* __restrict__ ids, int v) {
    int lo = 0, hi = Tn;
    while (lo < hi) {
        int mid = (lo + hi) >> 1;
        if (ids[mid] < v) lo = mid + 1; else hi = mid;
    }
    return lo;
}

__global__ __launch_bounds__(256)
void segmean_wmma_kernel(const float* __restrict__ emb,
                         const int*   __restrict__ wids,
                         float*       __restrict__ out) {
    const int b      = blockIdx.x / (Wn / WPB);
    const int wchunk = blockIdx.x % (Wn / WPB);
    const int w0     = wchunk * WPB;

    const int* __restrict__ ids = wids + (size_t)b * Tn;

    __shared__ int   s_bnd[WPB + 1];
    __shared__ float s_inv[WPB];

    const int tid = threadIdx.x;

    // Word span boundaries for words [w0, w0+16]: lower_bound over sorted ids.
    if (tid <= WPB) s_bnd[tid] = lbound4096(ids, w0 + tid);
    __syncthreads();

    if (tid < WPB) {
        int c = s_bnd[tid + 1] - s_bnd[tid];
        s_inv[tid] = 1.0f / (float)(c > 0 ? c : 1);
        // attention mask output (floats 0/1), laid out after emb_words
        float* att = out + (size_t)Bn * Wn * Dn;
        att[(size_t)b * Wn + w0 + tid] = (c > 0) ? 1.0f : 0.0f;
    }
    __syncthreads();

    const int s = s_bnd[0];
    const int e = s_bnd[WPB];

    const int lane   = tid & 31;    // wave32
    const int wave   = tid >> 5;    // 0..7, each owns 96 dims
    const int laneN  = lane & 15;
    const int hiHalf = lane >> 4;   // 0: K={0,1} / M=r ; 1: K={2,3} / M=r+8
    const int m      = laneN;       // word row this lane contributes to in A
    const float invm = s_inv[m];
    const int   wm   = w0 + m;

    v8f acc[6] = {};                // 6 N-tiles x 8 VGPRs of f32 means

    const float* __restrict__ embB = emb + (size_t)b * Tn * Dn;

    for (int k0 = s; k0 < e; k0 += 4) {
        const int kA = k0 + 2 * hiHalf;
        // Clamp addresses in-bounds and load unconditionally; ownership (A)
        // is zeroed for clamped/OOB tokens and for tokens of other word
        // tiles (ids sorted => id != wm), so B may safely load "don't care"
        // in-bounds data.  Bitwise & (not &&) keeps this branch-free.
        const int inb0 = (int)(kA     < Tn);
        const int inb1 = (int)(kA + 1 < Tn);
        const int i0 = inb0 ? kA     : (Tn - 1);
        const int i1 = inb1 ? kA + 1 : (Tn - 1);
        const int id0 = ids[i0];
        const int id1 = ids[i1];

        v2f A;
        A.x = (inb0 & (int)(id0 == wm)) ? invm : 0.0f;
        A.y = (inb1 & (int)(id1 == wm)) ? invm : 0.0f;

        const float* __restrict__ row0 = embB + (size_t)i0 * Dn;
        const float* __restrict__ row1 = embB + (size_t)i1 * Dn;

#pragma unroll
        for (int t = 0; t < 6; ++t) {
            const int d = wave * DPW + t * 16 + laneN;
            v2f Bv;
            Bv.x = __builtin_nontemporal_load(row0 + d);   // one-touch stream
            Bv.y = __builtin_nontemporal_load(row1 + d);
            // D = A x B + C : v_wmma_f32_16x16x4_f32
            acc[t] = __builtin_amdgcn_wmma_f32_16x16x4_f32(
                /*neg_a=*/false, A, /*neg_b=*/false, Bv,
                /*c_mod=*/(short)0, acc[t],
                /*reuse_a=*/false, /*reuse_b=*/false);
        }
    }

    // Store: C/D tile layout — VGPR r: lanes0-15 (M=r), lanes16-31 (M=r+8).
    // Write-once output -> non-temporal stores, keep L2 for word_ids.
#pragma unroll
    for (int t = 0; t < 6; ++t) {
        const int d = wave * DPW + t * 16 + laneN;
        size_t base = ((size_t)b * Wn + w0 + 8 * hiHalf) * Dn + d;
#pragma unroll
        for (int r = 0; r < 8; ++r) {
            __builtin_nontemporal_store(acc[t][r], out + base + (size_t)r * Dn);
        }
    }
}

extern "C" void kernel_launch(void* const* d_in, const int* in_sizes, int n_in,
                              void* d_out, int out_size, void* d_ws, size_t ws_size,
                              hipStream_t stream) {
    const float* emb  = (const float*)d_in[0];   // [B, T, D] f32
    const int*   wids = (const int*)d_in[1];     // [B, T] i32, sorted per row
    float*       out  = (float*)d_out;           // [B*W*D] emb_words ++ [B*W] att

    dim3 grid(Bn * (Wn / WPB));   // 4096 blocks: one 16-word tile each
    dim3 block(256);              // 8 wave32s, 96 dims per wave
    segmean_wmma_kernel<<<grid, block, 0, stream>>>(emb, wids, out);
}